// PRGAT2_38379827757723
// MI455X (gfx1250) — compile-verified
//
#include <hip/hip_runtime.h>
#include <math.h>

// ---------------------------------------------------------------------------
// RGCN -> GAT -> RGCN -> GAT pipeline for MI455X (gfx1250).
// Dense GEMMs use V_WMMA_F32_16X16X4_F32 (fp32 matrix pipe, wave32),
// branch-free inner loop, 4 WMMA tiles per wave reusing the B fragment.
// Graph scatter uses global float atomics; intermediates stay L2-resident.
// Workspace requirement: ~98 MB of d_ws.
// ---------------------------------------------------------------------------

typedef __attribute__((ext_vector_type(2))) float v2f;
typedef __attribute__((ext_vector_type(8))) float v8f;

#define NNODES   30000
#define NEDGES   480000
#define NRELS    8
#define NBASES   4
#define FIN      32
#define HIDDIM   64
#define NHEADS   4
#define NCLS     3
#define NEG_SLOPE 0.2f

// ---------------------------------------------------------------- utilities
__global__ void fill_kernel(float* __restrict__ p, float v, size_t n) {
  size_t i = (size_t)blockIdx.x * blockDim.x + threadIdx.x;
  if (i < n) p[i] = v;
}

__device__ __forceinline__ void atomicMaxFloat(float* addr, float val) {
  int* ai = (int*)addr;
  int old = __float_as_int(*addr);
  while (__int_as_float(old) < val) {
    int prev = atomicCAS(ai, old, __float_as_int(val));
    if (prev == old) break;
    old = prev;
  }
}

__device__ __forceinline__ float leaky(float v) {
  return v > 0.f ? v : NEG_SLOPE * v;
}
__device__ __forceinline__ float elu(float v) {
  return v > 0.f ? v : (expf(v) - 1.f);
}

// ------------------------------------------- w[r,i,o] = sum_b comp[r,b]*basis[b,i,o]
__global__ void basis_combine_kernel(const float* __restrict__ comp,
                                     const float* __restrict__ basis,
                                     float* __restrict__ w, int R, int B, int IO) {
  int i = blockIdx.x * blockDim.x + threadIdx.x;
  if (i >= R * IO) return;
  int r = i / IO, io = i % IO;
  float s = 0.f;
  for (int b = 0; b < B; ++b) s += comp[r * B + b] * basis[(size_t)b * IO + io];
  w[i] = s;
}

// ---------------------------------------------------------------------------
// fp32 GEMM:  out[M,N] = A[M,K](lda) * W[K,N](ldw), row-major, via
// V_WMMA_F32_16X16X4_F32.  Branch-free inner loop.
// Requirements: M % 16 == 0, N % 16 == 0, K % 4 == 0 (all call sites satisfy).
// Each wave computes MT (=4) vertically stacked 16x16 tiles of one 16-col
// panel: B fragment loaded once per k-step and reused by MT WMMAs; A fragment
// is one global_load_b64 per tile per k-step.  Row-blocks overhanging M clamp
// their A row (garbage computed, discarded at the uniform store guard).
//
// Fragment layouts (ISA 7.12.2):
//  A 16x4 f32 : lanes 0-15 -> K = k0,k0+1 ; lanes 16-31 -> K = k0+2,k0+3 (row = lane%16)
//  B  4x16 f32: mirrored (lane group selects K pair, lane%16 = column)
//  C/D 16x16  : VGPR v -> row v (lanes 0-15) / row v+8 (lanes 16-31), lane%16 = column
// ---------------------------------------------------------------------------
template <int MT>
__global__ void wmma_gemm_f32_kernel(const float* __restrict__ A,
                                     const float* __restrict__ W,
                                     float* __restrict__ out,
                                     int M, int N, int K, int lda, int ldw, int ldo) {
  const int lane = threadIdx.x & 31;
  const int wave = threadIdx.x >> 5;
  const int wavesPerBlock = blockDim.x >> 5;
  const int mr = lane & 15;
  const int kh = lane >> 4;                 // which K pair this lane holds

  const int col0 = blockIdx.x << 4;         // one 16-column panel per blockIdx.x
  const int rowBase = (blockIdx.y * wavesPerBlock + wave) * (MT * 16);
  if (rowBase >= M || col0 >= N) return;    // wave-uniform; EXEC stays all-1

  const int bcol = col0 + mr;

  const float* Ap[MT];
#pragma unroll
  for (int t = 0; t < MT; ++t) {
    int r = rowBase + t * 16 + mr;
    if (r > M - 1) r = M - 1;               // clamp: no predicated loads
    Ap[t] = A + (size_t)r * lda;
  }

  v8f acc[MT];
#pragma unroll
  for (int t = 0; t < MT; ++t) acc[t] = v8f{};

  for (int k0 = 0; k0 < K; k0 += 4) {
    const int ka = k0 + (kh << 1);
    v2f b;
    b.x = W[(size_t)ka * ldw + bcol];
    b.y = W[(size_t)(ka + 1) * ldw + bcol];
#pragma unroll
    for (int t = 0; t < MT; ++t) {
      v2f a = *(const v2f*)(Ap[t] + ka);    // 8B-aligned (lda, ka even)
      acc[t] = __builtin_amdgcn_wmma_f32_16x16x4_f32(false, a, false, b,
                                                     (short)0, acc[t], false, false);
    }
  }

#pragma unroll
  for (int t = 0; t < MT; ++t) {
    int row0 = rowBase + t * 16;
    if (row0 >= M) break;                   // uniform guard (M % 16 == 0)
    int srow = row0 + (kh << 3);
#pragma unroll
    for (int v = 0; v < 8; ++v)
      out[(size_t)(srow + v) * ldo + bcol] = acc[t][v];
  }
}

// plain GEMM for the tiny GAT2 projection (N = 12, not a multiple of 16)
__global__ void small_gemm_kernel(const float* __restrict__ A, const float* __restrict__ W,
                                  float* __restrict__ out, int M, int N, int K) {
  int i = blockIdx.x * blockDim.x + threadIdx.x;
  if (i >= M * N) return;
  int m = i / N, n = i % N;
  const float* a = A + (size_t)m * K;
  float s = 0.f;
  for (int k = 0; k < K; ++k) s += a[k] * W[(size_t)k * N + n];
  out[i] = s;
}

// ------------------------------------------------------------ RGCN pieces
__global__ void degree_kernel(const int* __restrict__ dst, float* __restrict__ deg, int E) {
  int e = blockIdx.x * blockDim.x + threadIdx.x;
  if (e < E) atomicAdd(&deg[dst[e]], 1.0f);
}

// agg[dst,:] += xr[src,:] for edges of relation `rel`; F = HIDDIM = 64
__global__ void rgcn_scatter_kernel(const int* __restrict__ src, const int* __restrict__ dst,
                                    const int* __restrict__ et, const float* __restrict__ xr,
                                    float* __restrict__ agg, int E, int rel) {
  size_t idx = (size_t)blockIdx.x * blockDim.x + threadIdx.x;
  size_t e = idx >> 6;
  int f = (int)(idx & 63);
  if (e >= (size_t)E) return;
  if (et[e] != rel) return;               // wave-uniform (32 lanes share one edge)
  atomicAdd(&agg[(size_t)dst[e] * HIDDIM + f], xr[(size_t)src[e] * HIDDIM + f]);
}

// out = elu(agg/max(deg,1) + rootc + bias)
__global__ void rgcn_combine_kernel(const float* __restrict__ agg, const float* __restrict__ deg,
                                    const float* __restrict__ rc, const float* __restrict__ bias,
                                    float* __restrict__ outp, int n, int F) {
  size_t i = (size_t)blockIdx.x * blockDim.x + threadIdx.x;
  if (i >= (size_t)n * F) return;
  int node = (int)(i / F), f = (int)(i % F);
  float v = agg[i] / fmaxf(deg[node], 1.0f) + rc[i] + bias[f];
  outp[i] = elu(v);
}

// ------------------------------------------------------------- GAT pieces
// a_s[n,h] = sum_c hg[n,h*C+c]*atts[h,c]; a_d likewise
__global__ void gat_scores_kernel(const float* __restrict__ hg, const float* __restrict__ atts,
                                  const float* __restrict__ attd, float* __restrict__ as_,
                                  float* __restrict__ ad_, int n, int H, int C) {
  int i = blockIdx.x * blockDim.x + threadIdx.x;
  if (i >= n * H) return;
  int node = i / H, h = i % H;
  const float* row = hg + (size_t)node * H * C + (size_t)h * C;
  float ss = 0.f, sd = 0.f;
  for (int c = 0; c < C; ++c) {
    ss += row[c] * atts[h * C + c];
    sd += row[c] * attd[h * C + c];
  }
  as_[i] = ss; ad_[i] = sd;
}

__global__ void gat_emax_kernel(const int* __restrict__ src, const int* __restrict__ dst,
                                const float* __restrict__ as_, const float* __restrict__ ad_,
                                float* __restrict__ emax, int E, int n, int H) {
  size_t idx = (size_t)blockIdx.x * blockDim.x + threadIdx.x;
  size_t tot = (size_t)(E + n) * H;
  if (idx >= tot) return;
  int h = (int)(idx % H); size_t e2 = idx / H;
  int s, d;
  if (e2 < (size_t)E) { s = src[e2]; d = dst[e2]; } else { s = d = (int)(e2 - E); }
  float v = leaky(as_[(size_t)s * H + h] + ad_[(size_t)d * H + h]);
  atomicMaxFloat(&emax[(size_t)d * H + h], v);
}

__global__ void gat_denom_kernel(const int* __restrict__ src, const int* __restrict__ dst,
                                 const float* __restrict__ as_, const float* __restrict__ ad_,
                                 const float* __restrict__ emax, float* __restrict__ den,
                                 int E, int n, int H) {
  size_t idx = (size_t)blockIdx.x * blockDim.x + threadIdx.x;
  size_t tot = (size_t)(E + n) * H;
  if (idx >= tot) return;
  int h = (int)(idx % H); size_t e2 = idx / H;
  int s, d;
  if (e2 < (size_t)E) { s = src[e2]; d = dst[e2]; } else { s = d = (int)(e2 - E); }
  float v = leaky(as_[(size_t)s * H + h] + ad_[(size_t)d * H + h]);
  atomicAdd(&den[(size_t)d * H + h], expf(v - emax[(size_t)d * H + h]));
}

__global__ void gat_aggregate_kernel(const int* __restrict__ src, const int* __restrict__ dst,
                                     const float* __restrict__ as_, const float* __restrict__ ad_,
                                     const float* __restrict__ emax, const float* __restrict__ den,
                                     const float* __restrict__ hg, float* __restrict__ outp,
                                     int E, int n, int H, int C) {
  size_t idx = (size_t)blockIdx.x * blockDim.x + threadIdx.x;
  size_t tot = (size_t)(E + n) * H * C;
  if (idx >= tot) return;
  int c = (int)(idx % C); size_t t = idx / C;
  int h = (int)(t % H); size_t e2 = t / H;
  int s, d;
  if (e2 < (size_t)E) { s = src[e2]; d = dst[e2]; } else { s = d = (int)(e2 - E); }
  float v = leaky(as_[(size_t)s * H + h] + ad_[(size_t)d * H + h]);
  float alpha = expf(v - emax[(size_t)d * H + h]) / den[(size_t)d * H + h];
  atomicAdd(&outp[((size_t)d * H + h) * C + c], alpha * hg[((size_t)s * H + h) * C + c]);
}

// in-place: p = elu(p + bias[i % F])
__global__ void bias_elu_kernel(float* __restrict__ p, const float* __restrict__ bias,
                                int n, int F) {
  size_t i = (size_t)blockIdx.x * blockDim.x + threadIdx.x;
  if (i >= (size_t)n * F) return;
  p[i] = elu(p[i] + bias[i % F]);
}

// mean over heads + bias, mean over classes, tanh -> out[n]
__global__ void gat2_final_kernel(const float* __restrict__ g2o, const float* __restrict__ bias,
                                  float* __restrict__ outp, int n) {
  int i = blockIdx.x * blockDim.x + threadIdx.x;
  if (i >= n) return;
  float m = 0.f;
  for (int c = 0; c < NCLS; ++c) {
    float s = 0.f;
    for (int h = 0; h < NHEADS; ++h) s += g2o[((size_t)i * NHEADS + h) * NCLS + c];
    m += s / (float)NHEADS + bias[c];
  }
  outp[i] = tanhf(m / (float)NCLS);
}

// ---------------------------------------------------------------------------
static inline unsigned cdiv(size_t a, unsigned b) { return (unsigned)((a + b - 1) / b); }

extern "C" void kernel_launch(void* const* d_in, const int* in_sizes, int n_in,
                              void* d_out, int out_size, void* d_ws, size_t ws_size,
                              hipStream_t stream) {
  (void)in_sizes; (void)n_in; (void)out_size; (void)ws_size;

  const float* x     = (const float*)d_in[0];
  const int*   ei    = (const int*)d_in[1];
  const int*   src   = ei;
  const int*   dst   = ei + NEDGES;
  const int*   et    = (const int*)d_in[2];
  const float* basis1 = (const float*)d_in[3];
  const float* comp1  = (const float*)d_in[4];
  const float* root1  = (const float*)d_in[5];
  const float* brg1   = (const float*)d_in[6];
  const float* Wg1    = (const float*)d_in[7];
  const float* atts1  = (const float*)d_in[8];
  const float* attd1  = (const float*)d_in[9];
  const float* bg1    = (const float*)d_in[10];
  const float* basis2 = (const float*)d_in[11];
  const float* comp2  = (const float*)d_in[12];
  const float* root2  = (const float*)d_in[13];
  const float* brg2   = (const float*)d_in[14];
  const float* Wg2    = (const float*)d_in[15];
  const float* atts2  = (const float*)d_in[16];
  const float* attd2  = (const float*)d_in[17];
  const float* bg2    = (const float*)d_in[18];
  float* out = (float*)d_out;

  // ---- workspace layout (floats); total ~24.4M floats (~98 MB) ----
  float* ws = (float*)d_ws;
  size_t o = 0;
  float* W1  = ws + o; o += (size_t)NRELS * FIN * HIDDIM;                 // 16384
  float* W2  = ws + o; o += (size_t)NRELS * NHEADS * HIDDIM * HIDDIM;     // 131072
  float* XR  = ws + o; o += (size_t)NNODES * HIDDIM;                      // per-rel x@w[r]
  float* AGG = ws + o; o += (size_t)NNODES * HIDDIM;
  float* RC  = ws + o; o += (size_t)NNODES * HIDDIM;                      // root contribution
  float* DEG = ws + o; o += (size_t)NNODES;
  float* H1  = ws + o; o += (size_t)NNODES * HIDDIM;                      // also reused as H3
  float* HG1 = ws + o; o += (size_t)NNODES * NHEADS * HIDDIM;
  float* H2  = ws + o; o += (size_t)NNODES * NHEADS * HIDDIM;             // GAT1 out
  float* AS  = ws + o; o += (size_t)NNODES * NHEADS;
  float* AD  = ws + o; o += (size_t)NNODES * NHEADS;
  float* EMX = ws + o; o += (size_t)NNODES * NHEADS;
  float* DEN = ws + o; o += (size_t)NNODES * NHEADS;
  float* HG2 = ws + o; o += (size_t)NNODES * NHEADS * NCLS;
  float* G2O = ws + o; o += (size_t)NNODES * NHEADS * NCLS;

  auto fill = [&](float* p, float v, size_t n) {
    fill_kernel<<<cdiv(n, 256), 256, 0, stream>>>(p, v, n);
  };
  // WMMA GEMM: requires M%16==0, N%16==0, K%4==0
  constexpr int MT = 4;            // 16x16 tiles per wave (stacked in M)
  constexpr int WPB = 4;           // waves per block
  auto gemm = [&](const float* A, const float* Wm, float* O,
                  int M, int N, int K, int lda, int ldw, int ldo) {
    dim3 grid((unsigned)(N / 16), cdiv((size_t)M, MT * 16 * WPB));
    wmma_gemm_f32_kernel<MT><<<grid, WPB * 32, 0, stream>>>(A, Wm, O, M, N, K, lda, ldw, ldo);
  };

  const size_t E2   = (size_t)NEDGES + NNODES;        // edges + self loops
  const float NEGINF = -3.402823466e38f;

  // ================= relation weights & degrees (shared) =================
  basis_combine_kernel<<<cdiv((size_t)NRELS * FIN * HIDDIM, 256), 256, 0, stream>>>(
      comp1, basis1, W1, NRELS, NBASES, FIN * HIDDIM);
  basis_combine_kernel<<<cdiv((size_t)NRELS * NHEADS * HIDDIM * HIDDIM, 256), 256, 0, stream>>>(
      comp2, basis2, W2, NRELS, NBASES, NHEADS * HIDDIM * HIDDIM);
  fill(DEG, 0.f, NNODES);
  degree_kernel<<<cdiv(NEDGES, 256), 256, 0, stream>>>(dst, DEG, NEDGES);

  // ================= RGCN layer 1 =================
  fill(AGG, 0.f, (size_t)NNODES * HIDDIM);
  gemm(x, root1, RC, NNODES, HIDDIM, FIN, FIN, HIDDIM, HIDDIM);
  for (int r = 0; r < NRELS; ++r) {
    gemm(x, W1 + (size_t)r * FIN * HIDDIM, XR, NNODES, HIDDIM, FIN, FIN, HIDDIM, HIDDIM);
    rgcn_scatter_kernel<<<cdiv((size_t)NEDGES * HIDDIM, 256), 256, 0, stream>>>(
        src, dst, et, XR, AGG, NEDGES, r);
  }
  rgcn_combine_kernel<<<cdiv((size_t)NNODES * HIDDIM, 256), 256, 0, stream>>>(
      AGG, DEG, RC, brg1, H1, NNODES, HIDDIM);

  // ================= GAT layer 1 (concat) =================
  gemm(H1, Wg1, HG1, NNODES, NHEADS * HIDDIM, HIDDIM, HIDDIM, NHEADS * HIDDIM, NHEADS * HIDDIM);
  gat_scores_kernel<<<cdiv((size_t)NNODES * NHEADS, 256), 256, 0, stream>>>(
      HG1, atts1, attd1, AS, AD, NNODES, NHEADS, HIDDIM);
  fill(EMX, NEGINF, (size_t)NNODES * NHEADS);
  fill(DEN, 0.f, (size_t)NNODES * NHEADS);
  fill(H2, 0.f, (size_t)NNODES * NHEADS * HIDDIM);
  gat_emax_kernel<<<cdiv(E2 * NHEADS, 256), 256, 0, stream>>>(
      src, dst, AS, AD, EMX, NEDGES, NNODES, NHEADS);
  gat_denom_kernel<<<cdiv(E2 * NHEADS, 256), 256, 0, stream>>>(
      src, dst, AS, AD, EMX, DEN, NEDGES, NNODES, NHEADS);
  gat_aggregate_kernel<<<cdiv(E2 * NHEADS * HIDDIM, 256), 256, 0, stream>>>(
      src, dst, AS, AD, EMX, DEN, HG1, H2, NEDGES, NNODES, NHEADS, HIDDIM);
  bias_elu_kernel<<<cdiv((size_t)NNODES * NHEADS * HIDDIM, 256), 256, 0, stream>>>(
      H2, bg1, NNODES, NHEADS * HIDDIM);

  // ================= RGCN layer 2 (in dim = 256) =================
  const int IN2 = NHEADS * HIDDIM;
  fill(AGG, 0.f, (size_t)NNODES * HIDDIM);
  gemm(H2, root2, RC, NNODES, HIDDIM, IN2, IN2, HIDDIM, HIDDIM);
  for (int r = 0; r < NRELS; ++r) {
    gemm(H2, W2 + (size_t)r * IN2 * HIDDIM, XR, NNODES, HIDDIM, IN2, IN2, HIDDIM, HIDDIM);
    rgcn_scatter_kernel<<<cdiv((size_t)NEDGES * HIDDIM, 256), 256, 0, stream>>>(
        src, dst, et, XR, AGG, NEDGES, r);
  }
  rgcn_combine_kernel<<<cdiv((size_t)NNODES * HIDDIM, 256), 256, 0, stream>>>(
      AGG, DEG, RC, brg2, H1 /* H3 reuses H1 */, NNODES, HIDDIM);

  // ================= GAT layer 2 (mean over heads) =================
  small_gemm_kernel<<<cdiv((size_t)NNODES * NHEADS * NCLS, 256), 256, 0, stream>>>(
      H1, Wg2, HG2, NNODES, NHEADS * NCLS, HIDDIM);
  gat_scores_kernel<<<cdiv((size_t)NNODES * NHEADS, 256), 256, 0, stream>>>(
      HG2, atts2, attd2, AS, AD, NNODES, NHEADS, NCLS);
  fill(EMX, NEGINF, (size_t)NNODES * NHEADS);
  fill(DEN, 0.f, (size_t)NNODES * NHEADS);
  fill(G2O, 0.f, (size_t)NNODES * NHEADS * NCLS);
  gat_emax_kernel<<<cdiv(E2 * NHEADS, 256), 256, 0, stream>>>(
      src, dst, AS, AD, EMX, NEDGES, NNODES, NHEADS);
  gat_denom_kernel<<<cdiv(E2 * NHEADS, 256), 256, 0, stream>>>(
      src, dst, AS, AD, EMX, DEN, NEDGES, NNODES, NHEADS);
  gat_aggregate_kernel<<<cdiv(E2 * NHEADS * NCLS, 256), 256, 0, stream>>>(
      src, dst, AS, AD, EMX, DEN, HG2, G2O, NEDGES, NNODES, NHEADS, NCLS);
  gat2_final_kernel<<<cdiv(NNODES, 256), 256, 0, stream>>>(G2O, bg2, out, NNODES);
}